// CenterLoss_6236292514255
// MI455X (gfx1250) — compile-verified
//
#include <hip/hip_runtime.h>
#include <hip/hip_bf16.h>

typedef __attribute__((ext_vector_type(2))) float v2f;
typedef __attribute__((ext_vector_type(8))) float v8f;

#define BATCH_N 32768
#define FEAT_N 512
#define ROWS_PER_WAVE 16
#define WAVES_PER_BLOCK 8
#define ROWS_PER_BLOCK (ROWS_PER_WAVE * WAVES_PER_BLOCK) // 128
#define NUM_BLOCKS (BATCH_N / ROWS_PER_BLOCK)            // 256

// Constant-index unrolled extract from the 8-VGPR accumulator (avoids scratch).
__device__ __forceinline__ float v8f_extract(v8f a, int i) {
    float r = a[0];
    r = (i == 1) ? a[1] : r;
    r = (i == 2) ? a[2] : r;
    r = (i == 3) ? a[3] : r;
    r = (i == 4) ? a[4] : r;
    r = (i == 5) ? a[5] : r;
    r = (i == 6) ? a[6] : r;
    r = (i == 7) ? a[7] : r;
    return r;
}

// Kernel 1: per-wave 16-row tile.  acc(16x16) += A(16x4) * A^T(4x16) over K=512,
// where A = X_tile - C_tile.  diag(acc) = ||x_i - c_i||^2.
__global__ __launch_bounds__(256) void center_loss_partials(
    const float* __restrict__ x, const int* __restrict__ labels,
    const float* __restrict__ centers, float* __restrict__ partials) {
    __shared__ float s_wave[WAVES_PER_BLOCK];

    const int tid  = threadIdx.x;
    const int lane = tid & 31;
    const int wv   = tid >> 5;
    const int r    = lane & 15;  // row within 16-row tile (M for A, N for B)
    const int h    = lane >> 4;  // lane-half selects K = {2h, 2h+1} per chunk
    const int row  = blockIdx.x * ROWS_PER_BLOCK + wv * ROWS_PER_WAVE + r;
    const int lab  = labels[row];

    const float* __restrict__ xrow = x + (size_t)row * FEAT_N + 2 * h;
    const float* __restrict__ crow = centers + (size_t)lab * FEAT_N + 2 * h;

    v8f acc = {0.f, 0.f, 0.f, 0.f, 0.f, 0.f, 0.f, 0.f};
#pragma unroll 8
    for (int k = 0; k < FEAT_N; k += 4) {
        v2f xv = *(const v2f*)(xrow + k);
        v2f cv = *(const v2f*)(crow + k);
        v2f dv = xv - cv;
        // D = A*B + C with A == B == (X - C) fragment: Gram matrix accumulate.
        acc = __builtin_amdgcn_wmma_f32_16x16x4_f32(
            /*neg_a=*/false, dv, /*neg_b=*/false, dv,
            /*c_mod=*/(short)0, acc, /*reuse_a=*/false, /*reuse_b=*/false);
    }

    // Diagonal: element i at (lane=i, vgpr=i) for i<8; (lane=i+16, vgpr=i-8) for i>=8.
    const bool lo = (lane < 8);
    const bool hi = (lane >= 24);
    const int  di = lo ? lane : (hi ? (lane - 24) : 0);
    float d = v8f_extract(acc, di);
    d = fminf(fmaxf(d, 1e-12f), 1e12f); // clamp like reference
    d = (lo || hi) ? d : 0.0f;

    // Deterministic wave32 tree reduction.
#pragma unroll
    for (int m = 16; m >= 1; m >>= 1) d += __shfl_xor(d, m, 32);

    if (lane == 0) s_wave[wv] = d;
    __syncthreads();
    if (tid == 0) {
        float s = 0.0f;
#pragma unroll
        for (int i = 0; i < WAVES_PER_BLOCK; ++i) s += s_wave[i];
        partials[blockIdx.x] = s;
    }
}

// Kernel 2: deterministic fixed-order reduction of the 256 block partials.
__global__ __launch_bounds__(256) void reduce_partials(
    const float* __restrict__ partials, float* __restrict__ out) {
    __shared__ float s[NUM_BLOCKS];
    const int tid = threadIdx.x;
    s[tid] = partials[tid];
    __syncthreads();
#pragma unroll
    for (int st = NUM_BLOCKS / 2; st > 0; st >>= 1) {
        if (tid < st) s[tid] += s[tid + st];
        __syncthreads();
    }
    if (tid == 0) out[0] = s[0] * (1.0f / (float)BATCH_N);
}

extern "C" void kernel_launch(void* const* d_in, const int* in_sizes, int n_in,
                              void* d_out, int out_size, void* d_ws, size_t ws_size,
                              hipStream_t stream) {
    const float* x       = (const float*)d_in[0];
    const int*   labels  = (const int*)d_in[1];
    const float* centers = (const float*)d_in[2];
    float*       out     = (float*)d_out;
    float*       partials = (float*)d_ws; // 256 floats of scratch

    center_loss_partials<<<NUM_BLOCKS, 256, 0, stream>>>(x, labels, centers, partials);
    reduce_partials<<<1, NUM_BLOCKS, 0, stream>>>(partials, out);
}